// FOPooling_57406532878944
// MI455X (gfx1250) — compile-verified
//
#include <hip/hip_runtime.h>
#include <stdint.h>

// ---------------------------------------------------------------------------
// QRNN fo-pooling for MI455X (gfx1250).
//   c_t = f_t * c_{t-1} + (1-f_t) * z_t ;  h_t = o_t * c_t
//   f,o = sigmoid(raw) = 0.5*tanh(0.5*raw)+0.5 ;  z = tanh(raw)   (v_tanh_f32)
//
// Memory-bound scan: ~268 MB traffic -> ~11.5 us floor @ 23.3 TB/s.
// Latency hiding via Tensor Data Mover: each stage = 3 TDM 2D-tile loads
// (16 timesteps x 128 channels per gate) into a deep LDS ring buffer
// (8 buffers = 192 KB/block, ~112 timesteps prefetched => ~44 MB in flight
// across 256 blocks), tracked with TENSORcnt.
// ---------------------------------------------------------------------------

#define T_SEQ     512
#define BATCH     32
#define HID       1024
#define SEG       128                  // channels per block (= blockDim.x)
#define S_STEPS   16                   // timesteps per stage
#define NBUF      8                    // LDS ring depth (stages)
#define NSTAGE    (T_SEQ / S_STEPS)    // 32
#define ROW3H     (3 * HID)            // 3072 floats per (t,b) row
#define YSTRIDE_T (BATCH * ROW3H)      // 98304 floats between timesteps
#define BH        (BATCH * HID)        // 32768

typedef unsigned int v4u __attribute__((ext_vector_type(4)));
typedef int          v4i __attribute__((ext_vector_type(4)));
typedef int          v8i __attribute__((ext_vector_type(8)));

__device__ __forceinline__ float tanh_f32(float x) {
  float r;
  asm("v_tanh_f32 %0, %1" : "=v"(r) : "v"(x));   // CDNA5 TRANS32 op
  return r;
}
__device__ __forceinline__ float sigmoid_f32(float x) {
  return fmaf(0.5f, tanh_f32(0.5f * x), 0.5f);
}

// One TDM 2D tile load: n_rows rows of row_elems contiguous f32, row stride
// row_stride_elems, from global byte addr gaddr into LDS byte addr lds_byte.
// D# packing per CDNA5 ISA ch.8 (group0 128b, group1 256b, group2/3 128b).
__device__ __forceinline__ void tdm_load_2d(uint32_t lds_byte, uint64_t gaddr,
                                            uint32_t row_elems, uint32_t n_rows,
                                            uint64_t row_stride_elems) {
  v4u g0;
  g0.x = 1u;                                     // count=1, user descriptor
  g0.y = lds_byte;                               // lds_addr[31:0]
  g0.z = (uint32_t)(gaddr & 0xFFFFFFFFu);        // global_addr[31:0]
  g0.w = (uint32_t)((gaddr >> 32) & 0x01FFFFFFu) // global_addr[56:32]
         | (2u << 30);                           // type = 2 ("image")

  v8i g1;
  g1[0] = (int)(2u << 16);                               // data_size=2 (4B), mask=0
  g1[1] = (int)((row_elems & 0xFFFFu) << 16);            // tensor_dim0[15:0]
  g1[2] = (int)((row_elems >> 16) |
                ((n_rows & 0xFFFFu) << 16));             // td0[31:16] | td1[15:0]
  g1[3] = (int)((n_rows >> 16) |
                ((row_elems & 0xFFFFu) << 16));          // td1[31:16] | tile_dim0
  g1[4] = (int)(n_rows & 0xFFFFu);                       // tile_dim1 | tile_dim2=0
  g1[5] = (int)(uint32_t)(row_stride_elems & 0xFFFFFFFFu);        // stride0 lo
  g1[6] = (int)(uint32_t)((row_stride_elems >> 32) & 0xFFFFu);    // stride0 hi
  g1[7] = 0;                                                       // stride1=0

  v4i g2 = {1, 1, 0, 0};   // tensor_dim2=1, tensor_dim3=1, unused
  v4i g3 = {0, 0, 0, 0};
  v8i gx = {0, 0, 0, 0, 0, 0, 0, 0};   // extra SGPR group (probe: zero-filled)

  __builtin_amdgcn_tensor_load_to_lds(g0, g1, g2, g3, gx, 0);
}

// Issue one stage: 3 gate tiles (F, Z, O) into ring buffer slot s % NBUF.
__device__ __forceinline__ void issue_stage(int s, uint32_t lds_base,
                                            uint64_t ybase_blk) {
  const int k = s % NBUF;
  const uint64_t t0_bytes = (uint64_t)s * S_STEPS * (uint64_t)YSTRIDE_T * 4ull;
#pragma unroll
  for (int m = 0; m < 3; ++m) {
    uint32_t lds = lds_base + (uint32_t)(((k * 3 + m) * S_STEPS) * SEG * 4);
    uint64_t ga  = ybase_blk + t0_bytes + (uint64_t)(m * HID) * 4ull;
    tdm_load_2d(lds, ga, SEG, S_STEPS, (uint64_t)YSTRIDE_T);
  }
}

__global__ void __launch_bounds__(SEG)
fo_pooling_tdm_kernel(const float* __restrict__ Y,
                      const float* __restrict__ init_c,
                      const float* __restrict__ init_h,
                      float* __restrict__ out) {
  extern __shared__ float smem[];   // NBUF*3*S_STEPS*SEG floats = 192 KB

  const int tid = threadIdx.x;
  const int b   = (int)blockIdx.x >> 3;          // batch index
  const int h0  = ((int)blockIdx.x & 7) * SEG;   // channel-chunk base
  const int gch = b * HID + h0 + tid;            // global channel id

  // Wave id as a *scalar* (SGPR) so the TDM region compiles to a real
  // scalar branch: TDM ignores EXEC, so EXEC-masking is not sufficient.
  const int wave0 = (__builtin_amdgcn_readfirstlane((int)(threadIdx.x >> 5)) == 0);

  // LDS byte offset: flat LDS addresses carry the offset in addr[31:0].
  const uint32_t lds_base = (uint32_t)(uintptr_t)(&smem[0]);
  const uint64_t ybase_blk =
      (uint64_t)(uintptr_t)Y + (uint64_t)(b * ROW3H + h0) * 4ull;

  // Prologue: prefetch NBUF-1 stages deep (3 TDM ops each; TENSORcnt <= 21).
  if (wave0) {
#pragma unroll
    for (int s = 0; s < NBUF - 1; ++s) issue_stage(s, lds_base, ybase_blk);
  }

  // h_0 row is init_h; carry starts at init_c.
  out[gch] = init_h[gch];
  float c  = init_c[gch];

  for (int s = 0; s < NSTAGE; ++s) {
    // Wait until stage s's 3 TDM ops landed (in-order completion per wave).
    // Immediates only: steady state leaves 3*(NBUF-2)=18; drain at the tail.
    {
      const int rem = NSTAGE - 1 - s;            // stages issued beyond s
      if      (rem >= 6) __builtin_amdgcn_s_wait_tensorcnt(18);
      else if (rem == 5) __builtin_amdgcn_s_wait_tensorcnt(15);
      else if (rem == 4) __builtin_amdgcn_s_wait_tensorcnt(12);
      else if (rem == 3) __builtin_amdgcn_s_wait_tensorcnt(9);
      else if (rem == 2) __builtin_amdgcn_s_wait_tensorcnt(6);
      else if (rem == 1) __builtin_amdgcn_s_wait_tensorcnt(3);
      else               __builtin_amdgcn_s_wait_tensorcnt(0);
    }
    __syncthreads();   // stage s visible to all waves; stage s-1 fully consumed

    // Prefetch stage s+NBUF-1 into the slot freed at the barrier above.
    if (wave0) {
      const int ps = s + NBUF - 1;
      if (ps < NSTAGE) issue_stage(ps, lds_base, ybase_blk);
    }

    // Consume stage s from LDS (lane i -> bank i: conflict-free).
    const int k = s % NBUF;
    const float* fbuf = &smem[((k * 3 + 0) * S_STEPS) * SEG];
    const float* zbuf = &smem[((k * 3 + 1) * S_STEPS) * SEG];
    const float* obuf = &smem[((k * 3 + 2) * S_STEPS) * SEG];
    float* orow = out + ((uint64_t)(s * S_STEPS) + 1u) * BH + gch;

#pragma unroll
    for (int j = 0; j < S_STEPS; ++j) {
      const float fr = fbuf[j * SEG + tid];
      const float zr = zbuf[j * SEG + tid];
      const float og = obuf[j * SEG + tid];
      const float f = sigmoid_f32(fr);
      const float z = tanh_f32(zr);
      const float o = sigmoid_f32(og);
      c = fmaf(f, c - z, z);                 // f*c + (1-f)*z
      orow[(uint64_t)j * BH] = o * c;        // h_{t+1 row}
    }
  }
}

extern "C" void kernel_launch(void* const* d_in, const int* in_sizes, int n_in,
                              void* d_out, int out_size, void* d_ws, size_t ws_size,
                              hipStream_t stream) {
  (void)in_sizes; (void)n_in; (void)out_size; (void)d_ws; (void)ws_size;
  const float* Y      = (const float*)d_in[0];
  const float* init_c = (const float*)d_in[1];
  const float* init_h = (const float*)d_in[2];
  float* out          = (float*)d_out;

  dim3 grid(BATCH * (HID / SEG));   // 256 blocks
  dim3 block(SEG);                  // 128 threads = 4 waves
  size_t shmem = (size_t)NBUF * 3 * S_STEPS * SEG * sizeof(float);  // 196608 B
  hipLaunchKernelGGL(fo_pooling_tdm_kernel, grid, block, shmem, stream,
                     Y, init_c, init_h, out);
}